// SEF_39376260169848
// MI455X (gfx1250) — compile-verified
//
#include <hip/hip_runtime.h>
#include <hip/hip_bf16.h>

typedef __attribute__((ext_vector_type(2))) float v2f;
typedef __attribute__((ext_vector_type(8))) float v8f;

// ---------------------------------------------------------------------------
// helpers
// ---------------------------------------------------------------------------
__device__ __forceinline__ float wave_reduce_sum(float v) {
  #pragma unroll
  for (int o = 16; o > 0; o >>= 1) v += __shfl_xor(v, o, 32);
  return v;
}

// ---------------------------------------------------------------------------
// K0: init — zero BN partial sums, set self-loop in-degree = 1.0
// ---------------------------------------------------------------------------
__global__ void init_kernel(float* __restrict__ stats, float* __restrict__ degb,
                            float* __restrict__ degf, int n) {
  int i = blockIdx.x * blockDim.x + threadIdx.x;
  if (i < 64) stats[i] = 0.0f;
  if (i < n) { degb[i] = 1.0f; degf[i] = 1.0f; }
}

// ---------------------------------------------------------------------------
// K1: y = x @ W1 + b1  (K=2), store pre-BN, accumulate column sum / sumsq
// block = 256 threads = 8 nodes x 32 cols
// ---------------------------------------------------------------------------
__global__ void enc1_kernel(const float* __restrict__ x, const float* __restrict__ w1,
                            const float* __restrict__ b1, float* __restrict__ hpre,
                            float* __restrict__ stats, int n) {
  __shared__ float red[256];
  const int tid = threadIdx.x;
  const int col = tid & 31;
  const int node = blockIdx.x * 8 + (tid >> 5);
  float y = 0.0f;
  if (node < n) {
    y = fmaf(x[(size_t)node * 2 + 0], w1[col],
        fmaf(x[(size_t)node * 2 + 1], w1[32 + col], b1[col]));
    hpre[(size_t)node * 32 + col] = y;
  }
  red[tid] = y;
  __syncthreads();
  #pragma unroll
  for (int s = 128; s >= 32; s >>= 1) {
    if (tid < s) red[tid] += red[tid + s];
    __syncthreads();
  }
  if (tid < 32) atomicAdd(&stats[col], red[tid]);
  __syncthreads();
  red[tid] = y * y;
  __syncthreads();
  #pragma unroll
  for (int s = 128; s >= 32; s >>= 1) {
    if (tid < s) red[tid] += red[tid + s];
    __syncthreads();
  }
  if (tid < 32) atomicAdd(&stats[32 + col], red[tid]);
}

// ---------------------------------------------------------------------------
// K2: fold BN to per-column scale/shift:  h = y*scale + shift
// ---------------------------------------------------------------------------
__global__ void bnstats_kernel(float* __restrict__ stats, const float* __restrict__ gamma,
                               const float* __restrict__ beta, float inv_n) {
  int c = threadIdx.x;             // 32 threads
  float mu  = stats[c] * inv_n;
  float var = stats[32 + c] * inv_n - mu * mu;
  float sc  = rsqrtf(var + 1e-5f) * gamma[c];
  stats[64 + c] = sc;              // scale
  stats[96 + c] = beta[c] - mu * sc;  // shift
}

// ---------------------------------------------------------------------------
// K3: emb = PReLU(BN(hpre)) @ W2 + b2   — V_WMMA_F32_16X16X4_F32, fp32 exact.
// 256-thread block = 8 independent waves, one 16-row tile each.
// A 16x4 layout: lanes 0-15 hold K={0,1}, lanes 16-31 hold K={2,3} (ISA 7.12.2).
// ---------------------------------------------------------------------------
__global__ __launch_bounds__(256)
void enc2_wmma_kernel(const float* __restrict__ hpre, const float* __restrict__ stats,
                      const float* __restrict__ prelu_a, const float* __restrict__ W,
                      const float* __restrict__ bias, float* __restrict__ emb) {
  const int lane = threadIdx.x & 31;
  const int half = lane >> 4;
  const int m = lane & 15;
  const int tile = blockIdx.x * 8 + (threadIdx.x >> 5);
  const int rowbase = tile * 16;
  const float slope = prelu_a[0];
  const float* scale = stats + 64;
  const float* shift = stats + 96;
  const float* arow = hpre + (size_t)(rowbase + m) * 32;

  const float b0 = bias[m];
  const float b1 = bias[m + 16];
  v8f acc0, acc1;
  #pragma unroll
  for (int r = 0; r < 8; ++r) { acc0[r] = b0; acc1[r] = b1; }

  #pragma unroll
  for (int k = 0; k < 8; ++k) {
    const int kk = 4 * k + 2 * half;
    v2f ain = *(const v2f*)(arow + kk);     // 8B-aligned pair -> b64 load
    float a0 = fmaf(ain.x, scale[kk],     shift[kk]);
    float a1 = fmaf(ain.y, scale[kk + 1], shift[kk + 1]);
    a0 = a0 > 0.0f ? a0 : slope * a0;
    a1 = a1 > 0.0f ? a1 : slope * a1;
    v2f av;  av.x  = a0;                    av.y  = a1;
    v2f bv0; bv0.x = W[kk * 32 + m];        bv0.y = W[(kk + 1) * 32 + m];
    v2f bv1; bv1.x = W[kk * 32 + m + 16];   bv1.y = W[(kk + 1) * 32 + m + 16];
    acc0 = __builtin_amdgcn_wmma_f32_16x16x4_f32(false, av, false, bv0, (short)0, acc0, false, false);
    acc1 = __builtin_amdgcn_wmma_f32_16x16x4_f32(false, av, false, bv1, (short)0, acc1, false, false);
  }
  #pragma unroll
  for (int r = 0; r < 8; ++r) {
    const int ro = rowbase + r + 8 * half;
    emb[(size_t)ro * 32 + m]      = acc0[r];
    emb[(size_t)ro * 32 + m + 16] = acc1[r];
  }
}

// ---------------------------------------------------------------------------
// K12/13: out = A[N,32] @ W[32,32] + b  (plain fp32 WMMA, same tiling)
// ---------------------------------------------------------------------------
__global__ __launch_bounds__(256)
void gemm32_wmma_kernel(const float* __restrict__ A, const float* __restrict__ W,
                        const float* __restrict__ bias, float* __restrict__ out) {
  const int lane = threadIdx.x & 31;
  const int half = lane >> 4;
  const int m = lane & 15;
  const int tile = blockIdx.x * 8 + (threadIdx.x >> 5);
  const int rowbase = tile * 16;
  const float* arow = A + (size_t)(rowbase + m) * 32;

  const float b0 = bias[m];
  const float b1 = bias[m + 16];
  v8f acc0, acc1;
  #pragma unroll
  for (int r = 0; r < 8; ++r) { acc0[r] = b0; acc1[r] = b1; }

  #pragma unroll
  for (int k = 0; k < 8; ++k) {
    const int kk = 4 * k + 2 * half;
    v2f av = *(const v2f*)(arow + kk);      // b64 load
    v2f bv0; bv0.x = W[kk * 32 + m];        bv0.y = W[(kk + 1) * 32 + m];
    v2f bv1; bv1.x = W[kk * 32 + m + 16];   bv1.y = W[(kk + 1) * 32 + m + 16];
    acc0 = __builtin_amdgcn_wmma_f32_16x16x4_f32(false, av, false, bv0, (short)0, acc0, false, false);
    acc1 = __builtin_amdgcn_wmma_f32_16x16x4_f32(false, av, false, bv1, (short)0, acc1, false, false);
  }
  #pragma unroll
  for (int r = 0; r < 8; ++r) {
    const int ro = rowbase + r + 8 * half;
    out[(size_t)ro * 32 + m]      = acc0[r];
    out[(size_t)ro * 32 + m + 16] = acc1[r];
  }
}

// ---------------------------------------------------------------------------
// K5/6: inv row norm — one wave per node, float4 (b128) streaming.
// d is a multiple of 128 (2048 / 512), rows are 16B aligned.
// ---------------------------------------------------------------------------
__global__ void rownorm_kernel(const float* __restrict__ feats, float* __restrict__ invn,
                               int n, int d) {
  int gt = blockIdx.x * blockDim.x + threadIdx.x;
  int node = gt >> 5, lane = gt & 31;
  if (node >= n) return;
  const float4* f = (const float4*)(feats + (size_t)node * d);
  const int nv = d >> 2;               // float4 count per row
  float s0 = 0.0f, s1 = 0.0f, s2 = 0.0f, s3 = 0.0f;
  for (int j = lane; j < nv; j += 32) {
    float4 v = f[j];
    s0 = fmaf(v.x, v.x, s0); s1 = fmaf(v.y, v.y, s1);
    s2 = fmaf(v.z, v.z, s2); s3 = fmaf(v.w, v.w, s3);
  }
  float s = wave_reduce_sum((s0 + s1) + (s2 + s3));
  if (lane == 0) invn[node] = rsqrtf(s + 1e-12f);
}

// ---------------------------------------------------------------------------
// K7/8: per-edge cosine sim + weighted in-degree. Wave per edge, float4
// streaming; both rows come from the 192MB L2 (feature matrices resident).
// ---------------------------------------------------------------------------
__global__ void edgesim_kernel(const float* __restrict__ feats, const float* __restrict__ invn,
                               const int* __restrict__ src, const int* __restrict__ dst,
                               float* __restrict__ sim, float* __restrict__ deg, int e, int d) {
  int gt = blockIdx.x * blockDim.x + threadIdx.x;
  int w = gt >> 5, lane = gt & 31;
  if (w >= e) return;
  int s0 = src[w], d0 = dst[w];
  const float4* fs = (const float4*)(feats + (size_t)s0 * d);
  const float4* fd = (const float4*)(feats + (size_t)d0 * d);
  const int nv = d >> 2;
  float a0 = 0.0f, a1 = 0.0f, a2 = 0.0f, a3 = 0.0f;
  for (int j = lane; j < nv; j += 32) {
    float4 a = fs[j];
    float4 b = fd[j];
    a0 = fmaf(a.x, b.x, a0); a1 = fmaf(a.y, b.y, a1);
    a2 = fmaf(a.z, b.z, a2); a3 = fmaf(a.w, b.w, a3);
  }
  float acc = wave_reduce_sum((a0 + a1) + (a2 + a3));
  if (lane == 0) {
    float wv = acc * invn[s0] * invn[d0];
    wv = wv > 0.0f ? wv : 0.0f;           // ReLU
    sim[w] = wv;
    atomicAdd(&deg[d0], wv);
  }
}

// ---------------------------------------------------------------------------
// K9/10: dn = rsqrt(max(deg,1e-6)); agg = dn^2 * emb (self loop). Wave/node.
// deg is overwritten with dn (lane 0 writes after all lanes read).
// ---------------------------------------------------------------------------
__global__ void selfloop_kernel(float* __restrict__ deg, const float* __restrict__ emb,
                                float* __restrict__ agg, int n) {
  int gt = blockIdx.x * blockDim.x + threadIdx.x;
  int node = gt >> 5, lane = gt & 31;
  if (node >= n) return;
  float dn = rsqrtf(fmaxf(deg[node], 1e-6f));
  agg[(size_t)node * 32 + lane] = dn * dn * emb[(size_t)node * 32 + lane];
  if (lane == 0) deg[node] = dn;
}

// ---------------------------------------------------------------------------
// K10/11: edge scatter — agg[dst] += dn[src]*dn[dst]*sim * emb[src]
// Wave per edge, lane = channel -> coalesced fp32 global atomics.
// ---------------------------------------------------------------------------
__global__ void scatter_kernel(const int* __restrict__ src, const int* __restrict__ dst,
                               const float* __restrict__ sim, const float* __restrict__ dn,
                               const float* __restrict__ emb, float* __restrict__ agg, int e) {
  int gt = blockIdx.x * blockDim.x + threadIdx.x;
  int w = gt >> 5, lane = gt & 31;
  if (w >= e) return;
  int s0 = src[w], d0 = dst[w];
  float coef = dn[s0] * dn[d0] * sim[w];
  atomicAdd(&agg[(size_t)d0 * 32 + lane], coef * emb[(size_t)s0 * 32 + lane]);
}

// ---------------------------------------------------------------------------
// K14: out = emb.w0 + xb.wb + xf.wf + biases.  Wave per node.
// ---------------------------------------------------------------------------
__global__ void final_kernel(const float* __restrict__ emb, const float* __restrict__ xb,
                             const float* __restrict__ xf,
                             const float* __restrict__ w0w, const float* __restrict__ w0b,
                             const float* __restrict__ wbw, const float* __restrict__ wbb,
                             const float* __restrict__ wfw, const float* __restrict__ wfb,
                             float* __restrict__ out, int n) {
  int gt = blockIdx.x * blockDim.x + threadIdx.x;
  int node = gt >> 5, lane = gt & 31;
  if (node >= n) return;
  size_t base = (size_t)node * 32;
  float v = emb[base + lane] * w0w[lane] + xb[base + lane] * wbw[lane]
          + xf[base + lane] * wfw[lane];
  v = wave_reduce_sum(v);
  if (lane == 0) out[node] = v + w0b[0] + wbb[0] + wfb[0];
}

// ---------------------------------------------------------------------------
extern "C" void kernel_launch(void* const* d_in, const int* in_sizes, int n_in,
                              void* d_out, int out_size, void* d_ws, size_t ws_size,
                              hipStream_t stream) {
  const float* x     = (const float*)d_in[0];
  const float* bodyf = (const float*)d_in[1];
  const float* facef = (const float*)d_in[2];
  const int*   eb    = (const int*)d_in[3];
  const int*   ef    = (const int*)d_in[4];
  const float* w1    = (const float*)d_in[5];
  const float* b1    = (const float*)d_in[6];
  const float* gamma = (const float*)d_in[7];
  const float* beta  = (const float*)d_in[8];
  const float* pra   = (const float*)d_in[9];
  const float* w2    = (const float*)d_in[10];
  const float* b2    = (const float*)d_in[11];
  const float* w0w   = (const float*)d_in[12];
  const float* w0b   = (const float*)d_in[13];
  const float* gbw   = (const float*)d_in[14];
  const float* gbb   = (const float*)d_in[15];
  const float* gfw   = (const float*)d_in[16];
  const float* gfb   = (const float*)d_in[17];
  const float* wbw   = (const float*)d_in[18];
  const float* wbb   = (const float*)d_in[19];
  const float* wfw   = (const float*)d_in[20];
  const float* wfb   = (const float*)d_in[21];
  float* outp = (float*)d_out;

  const int n = in_sizes[0] / 2;       // 16384
  const int e = in_sizes[3] / 2;       // 131072
  const int dbody = in_sizes[1] / n;   // 2048
  const int dface = in_sizes[2] / n;   // 512

  float* ws    = (float*)d_ws;
  float* hpre  = ws;
  float* emb   = hpre + (size_t)n * 32;
  float* aggb  = emb  + (size_t)n * 32;
  float* aggf  = aggb + (size_t)n * 32;
  float* xb    = aggf + (size_t)n * 32;
  float* xf    = xb   + (size_t)n * 32;
  float* invnb = xf   + (size_t)n * 32;
  float* invnf = invnb + n;
  float* degb  = invnf + n;
  float* degf  = degb + n;
  float* simb  = degf + n;
  float* simf  = simb + e;
  float* stats = simf + e;             // 128 floats: sum, sumsq, scale, shift

  const int TB = 256;
  const int gNodeWave = (n * 32 + TB - 1) / TB;   // wave per node
  const int gEdgeWave = (e * 32 + TB - 1) / TB;   // wave per edge
  const int gTile     = (n / 16) / 8;             // 8 WMMA waves per block

  init_kernel<<<(n + TB - 1) / TB, TB, 0, stream>>>(stats, degb, degf, n);
  enc1_kernel<<<(n + 7) / 8, TB, 0, stream>>>(x, w1, b1, hpre, stats, n);
  bnstats_kernel<<<1, 32, 0, stream>>>(stats, gamma, beta, 1.0f / (float)n);
  enc2_wmma_kernel<<<gTile, TB, 0, stream>>>(hpre, stats, pra, w2, b2, emb);

  rownorm_kernel<<<gNodeWave, TB, 0, stream>>>(bodyf, invnb, n, dbody);
  rownorm_kernel<<<gNodeWave, TB, 0, stream>>>(facef, invnf, n, dface);

  edgesim_kernel<<<gEdgeWave, TB, 0, stream>>>(bodyf, invnb, eb, eb + e, simb, degb, e, dbody);
  edgesim_kernel<<<gEdgeWave, TB, 0, stream>>>(facef, invnf, ef, ef + e, simf, degf, e, dface);

  selfloop_kernel<<<gNodeWave, TB, 0, stream>>>(degb, emb, aggb, n);
  selfloop_kernel<<<gNodeWave, TB, 0, stream>>>(degf, emb, aggf, n);

  scatter_kernel<<<gEdgeWave, TB, 0, stream>>>(eb, eb + e, simb, degb, emb, aggb, e);
  scatter_kernel<<<gEdgeWave, TB, 0, stream>>>(ef, ef + e, simf, degf, emb, aggf, e);

  gemm32_wmma_kernel<<<gTile, TB, 0, stream>>>(aggb, gbw, gbb, xb);
  gemm32_wmma_kernel<<<gTile, TB, 0, stream>>>(aggf, gfw, gfb, xf);

  final_kernel<<<gNodeWave, TB, 0, stream>>>(emb, xb, xf, w0w, w0b, wbw, wbb,
                                             wfw, wfb, outp, n);
}